// Points_Sampler_70961449664641
// MI455X (gfx1250) — compile-verified
//
#include <hip/hip_runtime.h>
#include <stdint.h>

// FPS for B=8, N=32768, NPOINT=2048 on MI455X (gfx1250, wave32).
// One workgroup per batch. temp[] lives in VGPRs (32/thread, unrolled).
// Each wave leader speculatively async-DMAs its candidate xyz into LDS so the
// L2 fetch overlaps the cross-wave reduction (gfx1250 async-to-LDS path).

#define BATCH   8
#define NPTS    32768
#define NPOINT_ 2048
#define THREADS 1024
#define NWAVES  (THREADS / 32)
#define PPT     (NPTS / THREADS)   // 32 points per thread

__global__ __launch_bounds__(THREADS, 1)
void fps_kernel(const float* __restrict__ xyz, float* __restrict__ out) {
  const int b    = blockIdx.x;
  const int t    = threadIdx.x;
  const int lane = t & 31;
  const int wave = t >> 5;

  const float* __restrict__ pts  = xyz + (size_t)b * NPTS * 3;
  float* __restrict__       outb = out + (size_t)b * 3 * NPOINT_;

  __shared__ float s_red_v[NWAVES];
  __shared__ int   s_red_i[NWAVES];
  __shared__ float s_cand[NWAVES * 4];  // per-wave candidate xyz (16B slots)
  __shared__ int   s_win;               // wave id of the winning candidate

  // Running min-squared-distance, register resident (fully unrolled indexing).
  float temp[PPT];
#pragma unroll
  for (int k = 0; k < PPT; ++k) temp[k] = 1e10f;

  // First selection is index 0 (matches reference: idx[0] = 0).
  if (t == 0) {
    s_win     = 0;
    s_cand[0] = pts[0];
    s_cand[1] = pts[1];
    s_cand[2] = pts[2];
  }
  __syncthreads();

  // LDS byte address of this wave's candidate slot; SGPR-pair global base.
  const uint32_t lds_cand = (uint32_t)(uintptr_t)(&s_cand[4 * wave]);
  const uint64_t gbase    = (uint64_t)(uintptr_t)pts;

  for (int j = 0; j < NPOINT_; ++j) {
    const int   w  = s_win;
    const float lx = s_cand[4 * w + 0];
    const float ly = s_cand[4 * w + 1];
    const float lz = s_cand[4 * w + 2];

    // Emit the gathered output for the current selection: out[b][c][j].
    if (t == 0) {
      outb[j]               = lx;
      outb[NPOINT_ + j]     = ly;
      outb[2 * NPOINT_ + j] = lz;
    }

    // Update temp = min(temp, d2(last)) and track local argmax(temp).
    float bestv = -1.0f;
    int   besti = 0;
#pragma unroll
    for (int k = 0; k < PPT; ++k) {
      const int i = t + k * THREADS;                 // coalesced: 12B stride/lane
      const float* __restrict__ p = pts + i * 3;
      const float dx = p[0] - lx;
      const float dy = p[1] - ly;
      const float dz = p[2] - lz;
      const float d2 = dx * dx + dy * dy + dz * dz;  // same form as reference
      const float tm = fminf(temp[k], d2);
      temp[k] = tm;
      if (tm > bestv) { bestv = tm; besti = i; }     // strict > : first max wins
    }

    // wave32 butterfly reduction (lower index wins ties, like jnp.argmax).
#pragma unroll
    for (int off = 16; off > 0; off >>= 1) {
      const float ov = __shfl_xor(bestv, off, 32);
      const int   oi = __shfl_xor(besti, off, 32);
      if (ov > bestv || (ov == bestv && oi < besti)) { bestv = ov; besti = oi; }
    }

    if (lane == 0) {
      s_red_v[wave] = bestv;
      s_red_i[wave] = besti;
      // Speculatively async-DMA this wave's candidate xyz into its LDS slot.
      // These 32 DMAs fly while wave 0 does the cross-wave reduction below,
      // taking the L2 round-trip off the serial critical path.
      // (GVS mode: saddr base + 32-bit vgpr offset; INST_OFFSET applies to
      //  both the global and LDS side, so 0/4/8 land x,y,z in the slot.)
      const uint32_t goff = (uint32_t)besti * 12u;
      asm volatile(
          "global_load_async_to_lds_b32 %0, %1, %2\n\t"
          "global_load_async_to_lds_b32 %0, %1, %2 offset:4\n\t"
          "global_load_async_to_lds_b32 %0, %1, %2 offset:8"
          :
          : "v"(lds_cand), "v"(goff), "s"(gbase)
          : "memory");
    }
    __syncthreads();  // s_red_* visible; async DMAs still in flight

    // Cross-wave reduction in wave 0: pick winning (value, index, wave).
    if (wave == 0) {
      float v = s_red_v[lane];
      int   i = s_red_i[lane];
      int   ww = lane;
#pragma unroll
      for (int off = 16; off > 0; off >>= 1) {
        const float ov = __shfl_xor(v, off, 32);
        const int   oi = __shfl_xor(i, off, 32);
        const int   ow = __shfl_xor(ww, off, 32);
        if (ov > v || (ov == v && oi < i)) { v = ov; i = oi; ww = ow; }
      }
      if (lane == 0) s_win = ww;
    }

    // Drain each wave's own async DMA (overlapped with the reduction above),
    // then make the candidate slots + s_win visible to the whole workgroup.
    asm volatile("s_wait_asynccnt 0x0" ::: "memory");
    __syncthreads();
  }
}

extern "C" void kernel_launch(void* const* d_in, const int* in_sizes, int n_in,
                              void* d_out, int out_size, void* d_ws, size_t ws_size,
                              hipStream_t stream) {
  (void)in_sizes; (void)n_in; (void)d_ws; (void)ws_size; (void)out_size;
  const float* points_xyz = (const float*)d_in[0];   // (B, N, 3) float32
  // d_in[1] (points_xyz_t) and d_in[2] (features) are not needed: the gathered
  // output values are exactly the xyz coordinates of the selected indices.
  float* out = (float*)d_out;                        // (B, 3, NPOINT) float32
  fps_kernel<<<BATCH, THREADS, 0, stream>>>(points_xyz, out);
}